// GCN_33054068310403
// MI455X (gfx1250) — compile-verified
//
#include <hip/hip_runtime.h>

#define N_NODES 50000
#define N_EDGES 800000

typedef __attribute__((ext_vector_type(2))) float v2f;
typedef __attribute__((ext_vector_type(8))) float v8f;

// ---------------------------------------------------------------------------
// Degree / normalization kernels
// ---------------------------------------------------------------------------
__global__ void deg_init_kernel(float* __restrict__ deg, int n) {
    int i = blockIdx.x * blockDim.x + threadIdx.x;
    if (i < n) deg[i] = 1.0f;   // self loop contributes 1
}

__global__ void deg_accum_kernel(const long long* __restrict__ ei,
                                 float* __restrict__ deg, int E) {
    int e = blockIdx.x * blockDim.x + threadIdx.x;
    if (e < E) {
        int d = (int)ei[(size_t)E + e];
        atomicAdd(&deg[d], 1.0f);
    }
}

__global__ void deg_rsqrt_kernel(float* __restrict__ deg, int n) {
    int i = blockIdx.x * blockDim.x + threadIdx.x;
    if (i < n) deg[i] = rsqrtf(deg[i]);   // deg >= 1 always
}

// ---------------------------------------------------------------------------
// Repack W[K,NOUT] (row-major) into WMMA B-fragment-major layout, zero-padded
// to 16-column tiles:
//   Wp[((kk*NT + t)*32 + lane)*2 + j]
//     lane<16 : rows k0+0,k0+1 (j=0,1), col = t*16 + lane
//     lane>=16: rows k0+2,k0+3 (j=0,1), col = t*16 + lane-16
// so in the GEMM each lane reads its whole 4x16 fragment as ONE b64 load.
// ---------------------------------------------------------------------------
template<int K, int NOUT>
__global__ void repack_B_kernel(const float* __restrict__ W, float* __restrict__ Wp) {
    constexpr int NT = (NOUT + 15) / 16;
    constexpr int TOTAL = (K / 4) * NT * 64;
    int idx = blockIdx.x * blockDim.x + threadIdx.x;
    if (idx >= TOTAL) return;
    int j    = idx & 1;
    int lane = (idx >> 1) & 31;
    int rem  = idx >> 6;
    int t    = rem % NT;
    int kk   = rem / NT;
    int col  = t * 16 + (lane & 15);
    int row  = kk * 4 + 2 * (lane >> 4) + j;
    Wp[idx] = (col < NOUT) ? W[row * NOUT + col] : 0.0f;
}

// ---------------------------------------------------------------------------
// f32 WMMA GEMM: C[M,NOUT] = A[M,K] @ W[K,NOUT]  (W pre-repacked as Wp)
// Block = 256 threads (8 waves), one 16-row M tile staged in LDS.
// Each wave owns TPW adjacent 16x16 N tiles sharing one A fragment per k-step.
// All shapes compile-time: int-only address math, no guards in the hot loop.
// ---------------------------------------------------------------------------
template<int K, int NOUT>
__global__ __launch_bounds__(256)
void gemm_wmma_f32(const float* __restrict__ A, const float* __restrict__ Wp,
                   float* __restrict__ C) {
    constexpr int NT  = (NOUT + 15) / 16;   // 16-col tiles (16 or 3)
    constexpr int TPW = (NT + 7) / 8;       // tiles per wave  (2 or 1)
    constexpr int LDA = K + 4;              // padded: bank-conflict-free, 8B aligned
    __shared__ float As[16 * LDA];

    const int m0  = blockIdx.x * 16;
    const int tid = threadIdx.x;

    // Cooperative load of the 16 x K A tile into LDS (int math only).
    for (int idx = tid; idx < 16 * K; idx += 256) {
        int r = idx / K;
        int k = idx - r * K;
        As[r * LDA + k] = A[(m0 + r) * K + k];
    }
    __syncthreads();

    const int wave = tid >> 5;
    const int lane = tid & 31;
    const int half = lane >> 4;
    const int l15  = lane & 15;
    if (wave * TPW >= NT) return;           // wave-uniform; no barriers after this

    const v2f* __restrict__ Bp = (const v2f*)Wp;   // [K/4][NT][32] fragments
    const float* __restrict__ Arow = &As[l15 * LDA + 2 * half];

    v8f acc[TPW];
    #pragma unroll
    for (int j = 0; j < TPW; ++j) { v8f z = {}; acc[j] = z; }

    #pragma unroll 8
    for (int kk = 0; kk < K / 4; ++kk) {
        // A fragment: one aligned 8-byte LDS read per lane.
        v2f a = *(const v2f*)&Arow[kk * 4];
        #pragma unroll
        for (int j = 0; j < TPW; ++j) {
            const int t = wave * TPW + j;
            v2f b = Bp[(kk * NT + t) * 32 + lane];  // one global_load_b64
            acc[j] = __builtin_amdgcn_wmma_f32_16x16x4_f32(
                false, a, false, b, (short)0, acc[j], false, false);
        }
    }

    // C layout: VGPR i holds row M=i (lanes 0-15) / M=i+8 (lanes 16-31).
    #pragma unroll
    for (int j = 0; j < TPW; ++j) {
        const int t   = wave * TPW + j;
        const int col = t * 16 + l15;
        if ((NOUT % 16 == 0) || col < NOUT) {
            #pragma unroll
            for (int i = 0; i < 8; ++i) {
                int row = m0 + i + half * 8;
                C[row * NOUT + col] = acc[j][i];
            }
        }
    }
}

// ---------------------------------------------------------------------------
// agg[i,c] = h[i,c] * dis[i]^2   (self-loop term; also initializes the buffer)
// ---------------------------------------------------------------------------
template<int F>
__global__ void selfloop_init_kernel(const float* __restrict__ h,
                                     const float* __restrict__ dis,
                                     float* __restrict__ agg) {
    int idx = blockIdx.x * blockDim.x + threadIdx.x;   // < 12.8M, int is fine
    if (idx < N_NODES * F) {
        int i = idx / F;                                // constant divisor
        float d = dis[i];
        agg[idx] = h[idx] * d * d;
    }
}

// ---------------------------------------------------------------------------
// One wave per edge: agg[dst,:] += h[src,:] * dis[src]*dis[dst]
// Coalesced row reads (L2-resident h) + coalesced f32 atomics (L2-resident agg).
// ---------------------------------------------------------------------------
template<int F>
__global__ __launch_bounds__(256)
void edge_aggregate_kernel(const long long* __restrict__ ei,
                           const float* __restrict__ dis,
                           const float* __restrict__ h,
                           float* __restrict__ agg) {
    int wave = blockIdx.x * (blockDim.x >> 5) + (threadIdx.x >> 5);
    int lane = threadIdx.x & 31;
    if (wave >= N_EDGES) return;            // wave-uniform
    int s = (int)ei[wave];
    int d = (int)ei[(size_t)N_EDGES + wave];
    float norm = dis[s] * dis[d];
    const float* __restrict__ hrow = h + (size_t)s * F;
    float* __restrict__ arow = agg + (size_t)d * F;
    #pragma unroll
    for (int c0 = 0; c0 < F; c0 += 32) {
        int c = c0 + lane;
        if ((F % 32 == 0) || c < F)
            atomicAdd(&arow[c], hrow[c] * norm);
    }
}

// ---------------------------------------------------------------------------
// x[i,c] = act(x[i,c] + b[c])
// ---------------------------------------------------------------------------
template<int F, bool RELU>
__global__ void bias_act_kernel(float* __restrict__ x, const float* __restrict__ b) {
    int idx = blockIdx.x * blockDim.x + threadIdx.x;
    if (idx < N_NODES * F) {
        int c = idx % F;                                // constant divisor
        float v = x[idx] + b[c];
        x[idx] = RELU ? fmaxf(v, 0.0f) : v;
    }
}

// ---------------------------------------------------------------------------
extern "C" void kernel_launch(void* const* d_in, const int* in_sizes, int n_in,
                              void* d_out, int out_size, void* d_ws, size_t ws_size,
                              hipStream_t stream) {
    (void)in_sizes; (void)n_in; (void)out_size; (void)ws_size;

    const float*     x  = (const float*)d_in[0];
    const long long* ei = (const long long*)d_in[1];   // int64 [2, E]
    const float*     W1 = (const float*)d_in[2];
    const float*     b1 = (const float*)d_in[3];
    const float*     W2 = (const float*)d_in[4];
    const float*     b2 = (const float*)d_in[5];
    const float*     W3 = (const float*)d_in[6];
    const float*     b3 = (const float*)d_in[7];
    float*           out = (float*)d_out;

    // Workspace layout (bytes):
    //   dis  : 50000 f32                 @ 0
    //   buf0 : 50000 x 256 f32           @ 200704
    //   buf1 : 50000 x 256 f32           @ 51400704
    //   Wp1  : 32*16*64 f32 (131072 B)   @ 102600704
    //   Wp2  : 64*16*64 f32 (262144 B)   @ 102731776
    //   Wp3  : 64*3*64  f32 ( 49152 B)   @ 102993920
    char* ws = (char*)d_ws;
    float* dis  = (float*)(ws);
    float* buf0 = (float*)(ws + 200704);
    float* buf1 = (float*)(ws + 51400704);
    float* Wp1  = (float*)(ws + 102600704);
    float* Wp2  = (float*)(ws + 102731776);
    float* Wp3  = (float*)(ws + 102993920);

    const int TB = 256;
    const int gridNodes   = (N_NODES + TB - 1) / TB;        // 196
    const int gridEdges   = (N_EDGES + TB - 1) / TB;        // 3125
    const int gridEdgeWav = N_EDGES / 8;                    // 100000 (8 waves/block)
    const int gridElemW   = (N_NODES * 256) / TB;           // 50000
    const int gridElemO   = (N_NODES * 40 + TB - 1) / TB;   // 7813
    const int gridGemm    = N_NODES / 16;                   // 3125 (exact)

    // --- Repack weights into B-fragment-major layout (tiny; L2-resident after)
    repack_B_kernel<128, 256><<<(32 * 16 * 64) / TB, TB, 0, stream>>>(W1, Wp1);
    repack_B_kernel<256, 256><<<(64 * 16 * 64) / TB, TB, 0, stream>>>(W2, Wp2);
    repack_B_kernel<256,  40><<<(64 *  3 * 64) / TB, TB, 0, stream>>>(W3, Wp3);

    // --- GCN normalization: deg (with self loops) -> dis = rsqrt(deg)
    deg_init_kernel  <<<gridNodes, TB, 0, stream>>>(dis, N_NODES);
    deg_accum_kernel <<<gridEdges, TB, 0, stream>>>(ei, dis, N_EDGES);
    deg_rsqrt_kernel <<<gridNodes, TB, 0, stream>>>(dis, N_NODES);

    // --- Layer 1: h = x @ W1 ; agg = selfloop + edges ; relu(agg + b1)
    gemm_wmma_f32<128, 256>   <<<gridGemm,    TB, 0, stream>>>(x, Wp1, buf0);
    selfloop_init_kernel<256> <<<gridElemW,   TB, 0, stream>>>(buf0, dis, buf1);
    edge_aggregate_kernel<256><<<gridEdgeWav, TB, 0, stream>>>(ei, dis, buf0, buf1);
    bias_act_kernel<256, true><<<gridElemW,   TB, 0, stream>>>(buf1, b1);

    // --- Layer 2
    gemm_wmma_f32<256, 256>   <<<gridGemm,    TB, 0, stream>>>(buf1, Wp2, buf0);
    selfloop_init_kernel<256> <<<gridElemW,   TB, 0, stream>>>(buf0, dis, buf1);
    edge_aggregate_kernel<256><<<gridEdgeWav, TB, 0, stream>>>(ei, dis, buf0, buf1);
    bias_act_kernel<256, true><<<gridElemW,   TB, 0, stream>>>(buf1, b2);

    // --- Layer 3 (NOUT = 40, no relu) — aggregate straight into d_out
    gemm_wmma_f32<256, 40>    <<<gridGemm,    TB, 0, stream>>>(buf1, Wp3, buf0);
    selfloop_init_kernel<40>  <<<gridElemO,   TB, 0, stream>>>(buf0, dis, out);
    edge_aggregate_kernel<40> <<<gridEdgeWav, TB, 0, stream>>>(ei, dis, buf0, out);
    bias_act_kernel<40, false><<<gridElemO,   TB, 0, stream>>>(out, b3);
}